// LocationSensitiveAttention_81552839017083
// MI455X (gfx1250) — compile-verified
//
#include <hip/hip_runtime.h>
#include <hip/hip_bf16.h>
#include <math.h>

typedef __attribute__((ext_vector_type(2))) float v2f;
typedef __attribute__((ext_vector_type(8))) float v8f;
typedef __attribute__((ext_vector_type(4))) int   v4i;

#define B_   64
#define T_   2000
#define E_   512
#define QD_  1024
#define U_   128
#define F_   32
#define KW_  31

// main-kernel tiling
constexpr int TS  = 128;   // t-rows per workgroup (8 waves x 16 rows)
constexpr int KC  = 32;    // K chunk of the memory stream staged per buffer
constexpr int LDA = 36;    // sA row stride (floats): 16B-aligned rows, bank-spread

// LDS budget (floats): W_keys k-packed (256KB) + W2 k-packed + 2x sA + prev win
constexpr int SW_F   = 128 * 128 * 4;          // 65536
constexpr int SW2_F  = 8 * 128 * 4;            // 4096
constexpr int SA_F   = TS * LDA;               // 4608 per buffer
constexpr int LDS_F  = SW_F + SW2_F + 2 * SA_F + 160;

#if __has_builtin(__builtin_amdgcn_global_load_async_to_lds_b128)
#define HAVE_ASYNC_LDS 1
#else
#define HAVE_ASYNC_LDS 0
#endif

__device__ __forceinline__ void copy16_to_lds(const float* g, float* l) {
#if HAVE_ASYNC_LDS
  __builtin_amdgcn_global_load_async_to_lds_b128(
      (__attribute__((address_space(1))) v4i*)g,
      (__attribute__((address_space(3))) v4i*)l, 0, 0);
#else
  *(float4*)l = *(const float4*)g;
#endif
}

__device__ __forceinline__ void wait_async_all() {
#if HAVE_ASYNC_LDS
  asm volatile("s_wait_asynccnt 0x0" ::: "memory");
#endif
}

// -------- prologue 1: pqb[b][u] = query@W_query + b_a + conv_b@W_loc --------
__global__ void pqb_kernel(const float* __restrict__ query,
                           const float* __restrict__ Wq,
                           const float* __restrict__ conv_b,
                           const float* __restrict__ W_loc,
                           const float* __restrict__ b_a,
                           float* __restrict__ pqb) {
  const int b = blockIdx.x, u = threadIdx.x;
  float s = b_a[u];
  for (int f = 0; f < F_; ++f) s = fmaf(conv_b[f], W_loc[f * U_ + u], s);
  const float* q = query + (size_t)b * QD_;
  for (int e = 0; e < QD_; ++e) s = fmaf(q[e], Wq[(size_t)e * U_ + u], s);
  pqb[b * U_ + u] = s;
}

// -------- prologue 2: fold conv into W2[k][u] = sum_f conv_w[k,0,f]*W_loc[f,u]
__global__ void w2_kernel(const float* __restrict__ conv_w,
                          const float* __restrict__ W_loc,
                          float* __restrict__ W2) {
  const int k = blockIdx.x, u = threadIdx.x;   // k: 0..31 (row 31 zero pad)
  float s = 0.f;
  if (k < KW_)
    for (int f = 0; f < F_; ++f) s = fmaf(conv_w[k * F_ + f], W_loc[f * U_ + u], s);
  W2[k * U_ + u] = s;
}

// -------- main fused kernel: keys GEMM + location GEMM + tanh-energy --------
__global__ __launch_bounds__(256)
void energy_kernel(const float* __restrict__ memory,   // [B,T,E]
                   const float* __restrict__ Wkeys,    // [E,U]
                   const float* __restrict__ prev,     // [B,T]
                   const float* __restrict__ pqb,      // [B,U]
                   const float* __restrict__ W2,       // [32,U]
                   const float* __restrict__ v_a,      // [U]
                   float* __restrict__ energy) {       // [B,T]
  const int b    = blockIdx.y;
  const int t0   = blockIdx.x * TS;
  const int tid  = threadIdx.x;
  const int lane = tid & 31;
  const int wave = tid >> 5;
  const int tw   = wave << 4;           // wave's 16-row base within the strip
  const int r    = lane & 15;           // row/col index within fragment
  const int kh   = (lane >> 4) << 1;    // K sub-offset per half-wave (0 or 2)

  extern __shared__ float lds[];
  float* sW    = lds;                   // [128 ksrow][128 n] float4 (k-packed)
  float* sW2s  = sW + SW_F;             // [8 ksrow][128 n] float4 (k-packed)
  float* sA0   = sW2s + SW2_F;          // [TS][LDA] double buffer 0
  float* sA1   = sA0 + SA_F;            // [TS][LDA] double buffer 1
  float* sPrev = sA1 + SA_F;            // [160] prev-alignment window
  float* sAbuf[2] = {sA0, sA1};

  // kick off the async HBM stream for chunk 0 immediately
  {
    float* dst = sAbuf[0];
    #pragma unroll
    for (int j = 0; j < 4; ++j) {
      const int i   = tid + (j << 8);
      const int row = i >> 3;             // 8 float4 per row (KC=32)
      const int c4  = (i & 7) << 2;
      const int t   = t0 + row;
      if (t < T_)
        copy16_to_lds(&memory[((size_t)b * T_ + t) * E_ + c4],
                      &dst[row * LDA + c4]);
    }
  }

  // stage all of W_keys into LDS, k-packed: sW[(ksrow*128+n)*4 + q] = Wk[4*ksrow+q][n]
  for (int i = tid; i < 128 * 128; i += 256) {
    const int ksrow = i >> 7, n = i & 127;
    float4 v;
    v.x = Wkeys[(size_t)((ksrow << 2) + 0) * U_ + n];
    v.y = Wkeys[(size_t)((ksrow << 2) + 1) * U_ + n];
    v.z = Wkeys[(size_t)((ksrow << 2) + 2) * U_ + n];
    v.w = Wkeys[(size_t)((ksrow << 2) + 3) * U_ + n];
    *(float4*)&sW[(size_t)i << 2] = v;
  }
  // stage folded conv weights, same k-packing (row 31 already zeroed in W2)
  for (int i = tid; i < 8 * 128; i += 256) {
    const int ksrow = i >> 7, n = i & 127;
    float4 v;
    v.x = W2[((ksrow << 2) + 0) * U_ + n];
    v.y = W2[((ksrow << 2) + 1) * U_ + n];
    v.z = W2[((ksrow << 2) + 2) * U_ + n];
    v.w = W2[((ksrow << 2) + 3) * U_ + n];
    *(float4*)&sW2s[i << 2] = v;
  }
  // prev window with zero-padded 'SAME' edges
  for (int i = tid; i < 160; i += 256) {
    const int t = t0 - 15 + i;
    sPrev[i] = (t >= 0 && t < T_) ? prev[(size_t)b * T_ + t] : 0.f;
  }
  wait_async_all();
  __syncthreads();

  v8f acc[8] = {};   // 16 rows x 128 cols of C per wave

  // location term: [TS x 32(K)] x [32 x 128] via im2col window
  #pragma unroll
  for (int ks = 0; ks < 8; ++ks) {
    v2f a;
    a.x = sPrev[tw + r + (ks << 2) + kh];
    a.y = sPrev[tw + r + (ks << 2) + kh + 1];
    #pragma unroll
    for (int n = 0; n < 8; ++n) {
      const int col = (n << 4) + r;
      v2f bm = *(const v2f*)&sW2s[(((ks << 7) + col) << 2) + kh];
      acc[n] = __builtin_amdgcn_wmma_f32_16x16x4_f32(
          false, a, false, bm, (short)0, acc[n], false, false);
    }
  }

  // keys term: async double-buffered stream of memory[b, t0:t0+TS, :]
  for (int kc = 0; kc < E_ / KC; ++kc) {
    // prefetch next chunk while computing this one
    if (kc + 1 < E_ / KC) {
      float* dst = sAbuf[(kc + 1) & 1];
      const int kbase = (kc + 1) * KC;
      #pragma unroll
      for (int j = 0; j < 4; ++j) {
        const int i   = tid + (j << 8);
        const int row = i >> 3;
        const int c4  = (i & 7) << 2;
        const int t   = t0 + row;
        if (t < T_)
          copy16_to_lds(&memory[((size_t)b * T_ + t) * E_ + kbase + c4],
                        &dst[row * LDA + c4]);
      }
    }
    const float* sAc = sAbuf[kc & 1];
    const int kbR = (kc * KC) >> 2;          // global ksrow base of this chunk
    const float* aRow = &sAc[(tw + r) * LDA + kh];
    #pragma unroll
    for (int ks = 0; ks < KC / 4; ++ks) {
      v2f a = *(const v2f*)(aRow + (ks << 2));
      const float* wRow = &sW[((((kbR + ks) << 7) + r) << 2) + kh];
      #pragma unroll
      for (int n = 0; n < 8; ++n) {
        v2f bm = *(const v2f*)(wRow + (n << 6));   // ((n<<4)<<2) floats
        acc[n] = __builtin_amdgcn_wmma_f32_16x16x4_f32(
            false, a, false, bm, (short)0, acc[n], false, false);
      }
    }
    wait_async_all();
    __syncthreads();
  }

  // epilogue: energy[b][t] = sum_u v_a[u] * tanh(keys + pl + pqb)
  float va[8], pq[8];
  #pragma unroll
  for (int n = 0; n < 8; ++n) {
    const int u = (n << 4) + r;
    va[n] = v_a[u];
    pq[n] = pqb[b * U_ + u];
  }
  #pragma unroll
  for (int j = 0; j < 8; ++j) {
    float s = 0.f;
    #pragma unroll
    for (int n = 0; n < 8; ++n)
      s += va[n] * tanhf(acc[n][j] + pq[n]);
    #pragma unroll
    for (int m = 1; m < 16; m <<= 1)           // reduce over the 16 N-lanes
      s += __shfl_xor(s, m, 32);
    const int t = t0 + tw + ((lane >> 4) << 3) + j;  // M = j (+8 for hi half)
    if ((lane == 0 || lane == 16) && t < T_)
      energy[(size_t)b * T_ + t] = s;
  }
}

// -------- masked softmax + cumulative next_state --------
__global__ __launch_bounds__(256)
void softmax_kernel(const float* __restrict__ energy,
                    const float* __restrict__ prev,
                    const int* __restrict__ lens,
                    float* __restrict__ out) {   // [0,B*T): align; [B*T,2B*T): next
  const int b = blockIdx.x;
  const int tid = threadIdx.x;
  const int len = lens[b];
  __shared__ float sred[8];

  float mx = -3.0e38f;
  for (int t = tid; t < T_; t += 256) {
    const float e = (t < len) ? energy[(size_t)b * T_ + t] : -1e9f;
    mx = fmaxf(mx, e);
  }
  for (int m = 1; m < 32; m <<= 1) mx = fmaxf(mx, __shfl_xor(mx, m, 32));
  if ((tid & 31) == 0) sred[tid >> 5] = mx;
  __syncthreads();
  mx = sred[0];
  #pragma unroll
  for (int w = 1; w < 8; ++w) mx = fmaxf(mx, sred[w]);
  __syncthreads();

  float sum = 0.f;
  for (int t = tid; t < T_; t += 256) {
    const float e = (t < len) ? energy[(size_t)b * T_ + t] : -1e9f;
    sum += expf(e - mx);
  }
  for (int m = 1; m < 32; m <<= 1) sum += __shfl_xor(sum, m, 32);
  if ((tid & 31) == 0) sred[tid >> 5] = sum;
  __syncthreads();
  float tot = 0.f;
  #pragma unroll
  for (int w = 0; w < 8; ++w) tot += sred[w];
  const float inv = 1.f / tot;

  for (int t = tid; t < T_; t += 256) {
    const float e = (t < len) ? energy[(size_t)b * T_ + t] : -1e9f;
    const float a = expf(e - mx) * inv;
    out[(size_t)b * T_ + t] = a;
    out[(size_t)B_ * T_ + (size_t)b * T_ + t] = a + prev[(size_t)b * T_ + t];
  }
}

extern "C" void kernel_launch(void* const* d_in, const int* in_sizes, int n_in,
                              void* d_out, int out_size, void* d_ws, size_t ws_size,
                              hipStream_t stream) {
  const float* query  = (const float*)d_in[0];
  const float* prev   = (const float*)d_in[1];
  const float* memory = (const float*)d_in[2];
  const int*   lens   = (const int*)d_in[3];
  const float* Wq     = (const float*)d_in[4];
  const float* Wk     = (const float*)d_in[5];
  const float* convw  = (const float*)d_in[6];
  const float* convb  = (const float*)d_in[7];
  const float* Wloc   = (const float*)d_in[8];
  const float* va     = (const float*)d_in[9];
  const float* ba     = (const float*)d_in[10];
  float* out = (float*)d_out;

  float* pqb    = (float*)d_ws;          // [B,U]
  float* W2     = pqb + B_ * U_;         // [32,U] (row 31 zeroed)
  float* energy = W2 + 32 * U_;          // [B,T]

  pqb_kernel<<<B_, U_, 0, stream>>>(query, Wq, convb, Wloc, ba, pqb);
  w2_kernel<<<32, U_, 0, stream>>>(convw, Wloc, W2);

  const size_t ldsBytes = (size_t)LDS_F * sizeof(float);
  (void)hipFuncSetAttribute((const void*)energy_kernel,
                            hipFuncAttributeMaxDynamicSharedMemorySize,
                            (int)ldsBytes);
  energy_kernel<<<dim3((T_ + TS - 1) / TS, B_), 256, ldsBytes, stream>>>(
      memory, Wk, prev, pqb, W2, va, energy);

  softmax_kernel<<<B_, 256, 0, stream>>>(energy, prev, lens, out);
}